// Resampling_88854283419709
// MI455X (gfx1250) — compile-verified
//
#include <hip/hip_runtime.h>

// Problem constants (fixed by the reference): B=4, P=8, H=W=D=C=32
#define HWDC (32 * 32 * 32 * 32)   // elements per (b,p) slab

typedef __attribute__((ext_vector_type(2))) float v2f;
typedef __attribute__((ext_vector_type(8))) float v8f;

__device__ __forceinline__ float rdlane_f(float v, int lane) {
  return __int_as_float(__builtin_amdgcn_readlane(__float_as_int(v), lane));
}

__global__ __launch_bounds__(256) void resample3d_wmma_kernel(
    const float* __restrict__ fmap,   // (B,P,H,W,D,C) f32
    const float* __restrict__ theta,  // (B, P*12) f32
    float* __restrict__ out)          // (B,P,H,W,D,C) f32
{
  const int lane = threadIdx.x & 31;
  const int wid  = threadIdx.x >> 5;
  const int gw   = blockIdx.x * 8 + wid;   // global wave id
  const int bp   = gw >> 11;               // 2048 waves per (b,p)
  const int n0   = (gw & 2047) << 4;       // first of this wave's 16 points

  const int li   = lane & 15;              // row (A/D) or column (B/D) index
  const int half = lane >> 4;              // selects K pair {0,1} vs {2,3}

  // ---- A matrix: theta rows (3x4, zero-padded to 16x4) ----
  // Lane li holds A[li][half*2 + 0] in v0 and A[li][half*2 + 1] in v1.
  const float* tbp = theta + bp * 12;
  float ax = 0.0f, ay = 0.0f;
  if (li < 3) {
    ax = tbp[li * 4 + half * 2 + 0];
    ay = tbp[li * 4 + half * 2 + 1];
  }

  // ---- B matrix: homogeneous grid coords, 4x16 (rows x_t,y_t,z_t,1) ----
  // VGPR j, lanes 0-15 hold row K=j, lanes 16-31 hold row K=j+2, column=li.
  const int pt = n0 + li;                  // point (column) index
  const int h  = pt >> 10;                 // n = h*W*D + w*D + d
  const int w  = (pt >> 5) & 31;
  const int dd = pt & 31;
  const float inv = 2.0f / 31.0f;          // linspace(-1,1,32) step
  const float gx = fmaf((float)w,  inv, -1.0f);
  const float gy = fmaf((float)h,  inv, -1.0f);
  const float gz = fmaf((float)dd, inv, -1.0f);
  const float bx = half ? gz   : gx;       // rows 0,2
  const float by = half ? 1.0f : gy;       // rows 1,3

  v2f Amat = {ax, ay};
  v2f Bmat = {bx, by};
  v8f Cmat = {0.f, 0.f, 0.f, 0.f, 0.f, 0.f, 0.f, 0.f};
  // D = theta(3x4) x grid(4x16): D row0=x, row1=y, row2=z for 16 points.
  v8f Dmat = __builtin_amdgcn_wmma_f32_16x16x4_f32(
      /*neg_a=*/false, Amat, /*neg_b=*/false, Bmat,
      /*c_mod=*/(short)0, Cmat, /*reuse_a=*/false, /*reuse_b=*/false);

  const float* __restrict__ fbp = fmap + (size_t)bp * HWDC;
  float* __restrict__ obase = out + (size_t)bp * HWDC + (size_t)n0 * 32 + lane;

  // Lanes = channels (C == 32 == wave32). Each corner load is one fully
  // coalesced 128B wave access; fmap slab (134MB) is kept resident in the
  // 192MB L2 because output stores below are non-temporal.
#pragma unroll 4
  for (int i = 0; i < 16; ++i) {
    float x = rdlane_f(Dmat[0], i);  // D row 0, column i
    float y = rdlane_f(Dmat[1], i);
    float z = rdlane_f(Dmat[2], i);

    // reference: 0.5 * ((v + 1) * (max - 1)), max = 31 -> scale 15
    x = (x + 1.0f) * 15.0f;
    y = (y + 1.0f) * 15.0f;
    z = (z + 1.0f) * 15.0f;

    int x0 = (int)floorf(x), y0 = (int)floorf(y), z0 = (int)floorf(z);
    int x1 = x0 + 1, y1 = y0 + 1, z1 = z0 + 1;
    x0 = min(max(x0, 0), 31); x1 = min(max(x1, 0), 31);
    y0 = min(max(y0, 0), 31); y1 = min(max(y1, 0), 31);
    z0 = min(max(z0, 0), 31); z1 = min(max(z1, 0), 31);

    // weights use the clipped integer coords, exactly as the reference
    const float xd = x - (float)x0;
    const float yd = (float)y1 - y;
    const float zd = z - (float)z0;
    const float oxd = 1.0f - xd, oyd = 1.0f - yd, ozd = 1.0f - zd;

    // element offset within slab: y<<15 | x<<10 | z<<5 | lane
    const float* p_y1x0 = fbp + (y1 << 15) + (x0 << 10) + lane;
    const float* p_y0x0 = fbp + (y0 << 15) + (x0 << 10) + lane;
    const float* p_y1x1 = fbp + (y1 << 15) + (x1 << 10) + lane;
    const float* p_y0x1 = fbp + (y0 << 15) + (x1 << 10) + lane;

    const float c000 = p_y1x0[z0 << 5];
    const float c001 = p_y1x0[z1 << 5];
    const float c010 = p_y0x0[z0 << 5];
    const float c011 = p_y0x0[z1 << 5];
    const float c100 = p_y1x1[z0 << 5];
    const float c101 = p_y1x1[z1 << 5];
    const float c110 = p_y0x1[z0 << 5];
    const float c111 = p_y0x1[z1 << 5];

    float r;
    r = c000 * (oxd * oyd * ozd);
    r = fmaf(c100, xd  * oyd * ozd, r);
    r = fmaf(c010, oxd * yd  * ozd, r);
    r = fmaf(c001, oxd * oyd * zd,  r);
    r = fmaf(c101, xd  * oyd * zd,  r);
    r = fmaf(c011, oxd * yd  * zd,  r);
    r = fmaf(c110, xd  * yd  * ozd, r);
    r = fmaf(c111, xd  * yd  * zd,  r);

    // Output is write-once / never re-read: non-temporal store keeps the
    // gathered fmap resident in L2 instead of being evicted by the stream.
    __builtin_nontemporal_store(r, obase + i * 32);
  }
}

extern "C" void kernel_launch(void* const* d_in, const int* in_sizes, int n_in,
                              void* d_out, int out_size, void* d_ws, size_t ws_size,
                              hipStream_t stream) {
  (void)in_sizes; (void)n_in; (void)out_size; (void)d_ws; (void)ws_size;
  const float* fmap  = (const float*)d_in[0];  // (4,8,32,32,32,32) f32
  const float* theta = (const float*)d_in[1];  // (4,96) f32
  float* out = (float*)d_out;

  // 1,048,576 grid points / (8 waves * 16 points) = 8192 blocks of 256
  resample3d_wmma_kernel<<<dim3(8192), dim3(256), 0, stream>>>(fmap, theta, out);
}